// APPNPModel_36988258353248
// MI455X (gfx1250) — compile-verified
//
#include <hip/hip_runtime.h>

#define N_NODES  100000
#define E_EDGES  1600000
#define DIN      256
#define DOUT     64
#define K_STEPS  10
#define ALPHA    0.1f

typedef __attribute__((ext_vector_type(2))) float v2f;
typedef __attribute__((ext_vector_type(8))) float v8f;

// ---------------------------------------------------------------------------
// deg / norm helpers
// ---------------------------------------------------------------------------
__global__ void zero_f32_kernel(float* __restrict__ p, int n) {
    int i = blockIdx.x * blockDim.x + threadIdx.x;
    if (i < n) p[i] = 0.0f;
}

__global__ void deg_kernel(const int* __restrict__ dst, float* __restrict__ deg, int e) {
    int i = blockIdx.x * blockDim.x + threadIdx.x;
    if (i < e) atomicAdd(&deg[dst[i]], 1.0f);
}

// in-place: deg -> dinv = rsqrt(deg_in + 1)   (self-loop folded in; deg+1 >= 1)
__global__ void dinv_kernel(float* __restrict__ deg, int n) {
    int i = blockIdx.x * blockDim.x + threadIdx.x;
    if (i < n) deg[i] = rsqrtf(deg[i] + 1.0f);
}

// ---------------------------------------------------------------------------
// h = relu(x @ W^T + b) via V_WMMA_F32_16X16X4_F32
// One wave computes one 16x16 tile of h; K=256 consumed in 64 steps of 4.
// f32 16x16x4 lane layout (ISA 7.12.2):
//   A (16x4): lane L holds A[L%16][kb + 2*(L>>4) + {0,1}]      -> float2 load
//   B (4x16): lane L holds B[kb + 2*(L>>4) + {0,1}][L%16]
//             = W[nTile + L%16][kb + 2*(L>>4) + {0,1}]          -> float2 load
//   C/D:      lane L, vgpr r -> (M = r + 8*(L>>4), N = L%16)
// ---------------------------------------------------------------------------
__global__ void linear_relu_wmma_kernel(const float* __restrict__ x,
                                        const float* __restrict__ W,
                                        const float* __restrict__ bias,
                                        float* __restrict__ h) {
    const int lane  = threadIdx.x & 31;
    const int wave  = threadIdx.x >> 5;
    const int tile  = blockIdx.x * (blockDim.x >> 5) + wave;
    const int nTilesN = DOUT / 16;                 // 4
    const int mTile = (tile / nTilesN) * 16;       // node-row base
    const int nTile = (tile % nTilesN) * 16;       // feature-col base
    if (mTile >= N_NODES) return;

    const int r16   = lane & 15;
    const int khalf = (lane >> 4) * 2;             // 0 or 2

    const float* arow = x + (size_t)(mTile + r16) * DIN + khalf;
    const float* brow = W + (size_t)(nTile + r16) * DIN + khalf;

    v8f c = {};
    #pragma unroll 8
    for (int kb = 0; kb < DIN; kb += 4) {
        v2f a = *(const v2f*)(arow + kb);
        v2f b = *(const v2f*)(brow + kb);
        c = __builtin_amdgcn_wmma_f32_16x16x4_f32(
                /*neg_a=*/false, a, /*neg_b=*/false, b,
                /*c_mod=*/(short)0, c, /*reuse_a=*/false, /*reuse_b=*/false);
    }

    const int   ncol = nTile + r16;
    const float bv   = bias[ncol];
    #pragma unroll
    for (int r = 0; r < 8; ++r) {
        const int m = mTile + r + 8 * (lane >> 4);
        float v = c[r] + bv;
        h[(size_t)m * DOUT + ncol] = v > 0.0f ? v : 0.0f;
    }
}

// ---------------------------------------------------------------------------
// z_new = alpha*h + (1-alpha)*dinv[i]^2 * z        (teleport + self-loop edge,
// non-atomic; also serves as the accumulator init for the edge scatter)
// ---------------------------------------------------------------------------
__global__ void ppnp_init_kernel(const float* __restrict__ z,
                                 const float* __restrict__ h,
                                 const float* __restrict__ dinv,
                                 float* __restrict__ znew) {
    int idx = blockIdx.x * blockDim.x + threadIdx.x;
    if (idx < N_NODES * DOUT) {
        int node = idx / DOUT;
        float di = dinv[node];
        znew[idx] = ALPHA * h[idx] + (1.0f - ALPHA) * di * di * z[idx];
    }
}

// ---------------------------------------------------------------------------
// Edge scatter: one wave per edge, lane l handles features {2l, 2l+1}.
// Gather z[src] as coalesced float2 (256B/wave), scatter via f32 atomics.
// All of z / znew stays resident in the 192MB L2.
// ---------------------------------------------------------------------------
__global__ void ppnp_scatter_kernel(const int* __restrict__ src,
                                    const int* __restrict__ dst,
                                    const float* __restrict__ dinv,
                                    const float* __restrict__ z,
                                    float* __restrict__ znew) {
    long long t = (long long)blockIdx.x * blockDim.x + threadIdx.x;
    int e = (int)(t >> 5);
    if (e >= E_EDGES) return;
    int f = ((int)t & 31) * 2;

    int s = src[e];
    int d = dst[e];
    float w = (1.0f - ALPHA) * dinv[s] * dinv[d];

    v2f zv = *(const v2f*)(z + (size_t)s * DOUT + f);
    float* out = znew + (size_t)d * DOUT + f;
    atomicAdd(out + 0, w * zv.x);
    atomicAdd(out + 1, w * zv.y);
}

// ---------------------------------------------------------------------------
// launch
// ---------------------------------------------------------------------------
extern "C" void kernel_launch(void* const* d_in, const int* in_sizes, int n_in,
                              void* d_out, int out_size, void* d_ws, size_t ws_size,
                              hipStream_t stream) {
    const float* x  = (const float*)d_in[0];
    const int*   ei = (const int*)  d_in[1];   // (2, E): row 0 = src, row 1 = dst
    const float* W  = (const float*)d_in[2];
    const float* b  = (const float*)d_in[3];
    float*       out = (float*)d_out;

    const int* e_src = ei;
    const int* e_dst = ei + E_EDGES;

    // workspace layout
    float* dinv = (float*)d_ws;                       // N      (deg, then dinv)
    float* h    = dinv + N_NODES;                     // N*DOUT
    float* za   = h    + (size_t)N_NODES * DOUT;      // N*DOUT
    float* zb   = za   + (size_t)N_NODES * DOUT;      // N*DOUT

    const int NB256 = 256;

    // 1) degrees (with self-loop folded into rsqrt)
    zero_f32_kernel<<<(N_NODES + NB256 - 1) / NB256, NB256, 0, stream>>>(dinv, N_NODES);
    deg_kernel<<<(E_EDGES + NB256 - 1) / NB256, NB256, 0, stream>>>(e_dst, dinv, E_EDGES);
    dinv_kernel<<<(N_NODES + NB256 - 1) / NB256, NB256, 0, stream>>>(dinv, N_NODES);

    // 2) h = relu(x @ W^T + b): 6250 * 4 = 25000 tiles, 8 waves/block
    {
        const int tiles  = (N_NODES / 16) * (DOUT / 16);  // 25000
        const int blocks = tiles / 8;                     // 3125
        linear_relu_wmma_kernel<<<blocks, 256, 0, stream>>>(x, W, b, h);
    }

    // 3) K=10 propagation steps (ping-pong za/zb, final step lands in d_out)
    const int initBlocks    = (N_NODES * DOUT + NB256 - 1) / NB256;          // 25000
    const long long sthread = (long long)E_EDGES * 32;
    const int scatBlocks    = (int)((sthread + NB256 - 1) / NB256);          // 200000

    for (int k = 1; k <= K_STEPS; ++k) {
        const float* zsrc = (k == 1) ? h : ((k & 1) ? zb : za);
        float*       zdst = (k == K_STEPS) ? out : ((k & 1) ? za : zb);

        ppnp_init_kernel<<<initBlocks, NB256, 0, stream>>>(zsrc, h, dinv, zdst);
        ppnp_scatter_kernel<<<scatBlocks, NB256, 0, stream>>>(e_src, e_dst, dinv, zsrc, zdst);
    }
}